// DRR_26456998543708
// MI455X (gfx1250) — compile-verified
//
#include <hip/hip_runtime.h>
#include <math.h>

// ---------------------------------------------------------------------------
// DRR (Siddon ray casting) for MI455X / gfx1250.
//   - 1 thread = 1 ray (pixel). wave32 covers an 8x4 pixel patch -> coherent
//     gathers into the 64MB volume, which is fully L2-resident (192MB L2).
//   - 3-way merge of per-axis plane-crossing arithmetic progressions replaces
//     the reference's sort; fully branchless inner loop (v_cndmask) since the
//     axis-selection diverges per lane under wave32.
//   - Inner loop is software-pipelined (depth-2 pending): the merge state
//     never depends on loaded voxels, so each gather gets ~2 loop bodies of
//     latency slack (s_wait_loadcnt > 0) on top of multi-wave overlap.
//   - A prefetch kernel warms L2 with global_prefetch_b8 (gfx1250 path).
// ---------------------------------------------------------------------------

namespace {
constexpr int   HH   = 200;
constexpr int   WW   = 200;
constexpr int   NV   = 256;       // volume extent per axis
constexpr float DELX = 2.0f;
constexpr float DELY = 2.0f;
constexpr float EPSF = 1e-8f;
}

struct AxisState {
  float a;     // alpha of current (next) crossing; +inf when exhausted
  float fi;    // current plane index as float
  float d;     // d(alpha)/d(index) = spacing / sdd  (signed)
  float b;     // alpha at index 0 = -src / sdd
  float dirf;  // +1 / -1: index step direction for ascending alpha
  int   cnt;   // remaining crossings (including current)
};

__device__ __forceinline__ float axAlpha(float fi, float d, float b) {
  return fmaf(fi, d, b);
}

// Find the contiguous plane-index range whose alphas lie in [amin, amax],
// oriented so iteration goes in ascending alpha. Algebraic bounds + small
// fix-up so boundary inclusion matches the >= / <= tests exactly.
__device__ __forceinline__ void initAxis(AxisState& st, float amin, float amax) {
  const float d = st.d, b = st.b;
  const float FINF = __builtin_inff();

  float qlo, qhi;
  int dir;
  if (d > 0.0f) { qlo = (amin - b) / d; qhi = (amax - b) / d; dir = 1; }
  else          { qlo = (amax - b) / d; qhi = (amin - b) / d; dir = -1; }
  // clamp in float first (also neutralizes NaN) before int conversion
  qlo = fminf(fmaxf(qlo, -1.0f), (float)(NV + 1));
  qhi = fminf(fmaxf(qhi, -1.0f), (float)(NV + 1));
  int lo = (int)ceilf(qlo);
  int hi = (int)floorf(qhi);
  if (lo < 0)  lo = 0;
  if (hi > NV) hi = NV;

  int s, e;
  if (dir > 0) { s = lo; e = hi; } else { s = hi; e = lo; }
  int cnt = (e - s) * dir + 1;
  if (cnt < 0) cnt = 0;

  auto ok = [&](int i) -> bool {
    float a = axAlpha((float)i, d, b);
    return (a >= amin) && (a <= amax);
  };
  // shrink front/back to satisfy predicate (ulp-level fixups, bounded)
  while (cnt > 0 && !ok(s)) { s += dir; --cnt; }
  while (cnt > 0 && !ok(e)) { e -= dir; --cnt; }
  if (cnt > 0) {
    // expand by up to 2 on each side in case algebraic bound was too tight
    #pragma unroll
    for (int k = 0; k < 2; ++k) {
      int s2 = s - dir;
      if (s2 >= 0 && s2 <= NV && ok(s2)) { s = s2; ++cnt; }
    }
    #pragma unroll
    for (int k = 0; k < 2; ++k) {
      int e2 = e + dir;
      if (e2 >= 0 && e2 <= NV && ok(e2)) { e = e2; ++cnt; }
    }
  }

  st.dirf = (float)dir;
  st.cnt  = cnt;
  st.fi   = (float)s;
  st.a    = (cnt > 0) ? axAlpha((float)s, d, b) : FINF;
}

// Branchless advance of every axis whose current alpha equals the merged min.
__device__ __forceinline__ void advAxis(AxisState& S, float m) {
  const float FINF = __builtin_inff();
  bool  hit  = (S.a <= m);                 // S.a >= m always; <= means tie with min
  int   ncnt = S.cnt - (hit ? 1 : 0);
  float nfi  = S.fi + (hit ? S.dirf : 0.0f);
  float na   = axAlpha(nfi, S.d, S.b);
  S.cnt = ncnt;
  S.fi  = nfi;
  S.a   = hit ? ((ncnt > 0) ? na : FINF) : S.a;
}

// Warm the 192MB L2 with the 64MB volume: one global_prefetch_b8 per 128B line.
__global__ void __launch_bounds__(256) prefetch_l2(const float* __restrict__ p, int n) {
  int idx = (blockIdx.x * blockDim.x + threadIdx.x) << 5;   // 32 floats = 128B
  if (idx < n) {
    __builtin_prefetch(p + idx, 0, 3);
  }
}

__global__ void __launch_bounds__(64) drr_siddon(
    const float* __restrict__ vol,      // 256^3, x-major (x,y,z)
    const float* __restrict__ spacing,  // 3
    const float* __restrict__ sdrp,
    const float* __restrict__ thetap,
    const float* __restrict__ phip,
    const float* __restrict__ gammap,
    const float* __restrict__ bxp,
    const float* __restrict__ byp,
    const float* __restrict__ bzp,
    float* __restrict__ out)
{
  const float FINF = __builtin_inff();

  // 8x8 pixel tile per 64-thread block -> each wave32 = 8x4 pixel patch
  const int j = blockIdx.x * 8 + (threadIdx.x & 7);   // W (s / v) axis
  const int i = blockIdx.y * 8 + (threadIdx.x >> 3);  // H (t / u) axis
  if (i >= HH || j >= WW) return;

  // ---- uniform ray-geometry setup (scalar loads; trig once per thread) ----
  const float sdr = sdrp[0];
  const float th  = thetap[0], ph = phip[0], gm = gammap[0];
  const float spx = spacing[0], spy = spacing[1], spz = spacing[2];

  float sth, cth, sph, cph, sgm, cgm;
  sincosf(th, &sth, &cth);
  sincosf(ph, &sph, &cph);
  sincosf(gm, &sgm, &cgm);

  // R = sdr * Rz(theta)Ry(phi)Rx(gamma); source = R[:,0]; u,v = cols 1,2 of M
  const float s0x = sdr * (cth * cph);
  const float s0y = sdr * (sth * cph);
  const float s0z = sdr * (-sph);
  const float ux = cth * sph * sgm - sth * cgm;
  const float uy = sth * sph * sgm + cth * cgm;
  const float uz = cph * sgm;
  const float vx = cth * sph * cgm + sth * sgm;
  const float vy = sth * sph * cgm - cth * sgm;
  const float vz = cph * cgm;

  const float srcx = s0x + bxp[0];
  const float srcy = s0y + byp[0];
  const float srcz = s0z + bzp[0];

  const float t = (float)(i - 99) * DELX;   // (arange(-100,100)+1)*2
  const float s = (float)(j - 99) * DELY;

  // sdd = tgt - src + eps = t*u + s*v - 2*source + eps
  const float sddx = fmaf(t, ux, fmaf(s, vx, -2.0f * s0x)) + EPSF;
  const float sddy = fmaf(t, uy, fmaf(s, vy, -2.0f * s0y)) + EPSF;
  const float sddz = fmaf(t, uz, fmaf(s, vz, -2.0f * s0z)) + EPSF;

  // ---- per-axis crossing progressions: alpha(idx) = idx*d + b ----
  AxisState X, Y, Z;
  {
    const float ix = 1.0f / sddx, iy = 1.0f / sddy, iz = 1.0f / sddz;
    X.d = spx * ix; X.b = -srcx * ix;
    Y.d = spy * iy; Y.b = -srcy * iy;
    Z.d = spz * iz; Z.b = -srcz * iz;
  }
  const float fNV = (float)NV;
  float amin = -FINF, amax = FINF;
  {
    float a0 = X.b, a1 = axAlpha(fNV, X.d, X.b);
    amin = fmaxf(amin, fminf(a0, a1)); amax = fminf(amax, fmaxf(a0, a1));
    a0 = Y.b; a1 = axAlpha(fNV, Y.d, Y.b);
    amin = fmaxf(amin, fminf(a0, a1)); amax = fminf(amax, fmaxf(a0, a1));
    a0 = Z.b; a1 = axAlpha(fNV, Z.d, Z.b);
    amin = fmaxf(amin, fminf(a0, a1)); amax = fminf(amax, fmaxf(a0, a1));
  }
  initAxis(X, amin, amax);
  initAxis(Y, amin, amax);
  initAxis(Z, amin, amax);

  const float isx = 1.0f / spx, isy = 1.0f / spy, isz = 1.0f / spz;

  // ---- 3-way merge == sorted union of in-range crossings ----
  // Software pipeline, depth-2 pending: slot0 (oldest) and slot1. Primed with
  // zero-contribution entries (fmaf(0,0,acc) == acc exactly), drained after.
  float acc = 0.0f;
  float aprev = fminf(X.a, fminf(Y.a, Z.a));
  if (aprev < FINF) {
    advAxis(X, aprev); advAxis(Y, aprev); advAxis(Z, aprev);

    float s0 = 0.0f, v0 = 0.0f;   // pending slot 0 (consumed this iter)
    float s1 = 0.0f, v1 = 0.0f;   // pending slot 1

    for (;;) {
      const float m = fminf(X.a, fminf(Y.a, Z.a));
      if (!(m < FINF)) break;

      const float amid = 0.5f * (aprev + m);
      const float step = m - aprev;

      // sample point -> voxel index (trunc toward zero, clip to [0,255])
      float pxs = fmaf(amid, sddx, srcx) * isx;
      float pys = fmaf(amid, sddy, srcy) * isy;
      float pzs = fmaf(amid, sddz, srcz) * isz;
      pxs = fminf(fmaxf(pxs, 0.0f), (float)(NV - 1));
      pys = fminf(fmaxf(pys, 0.0f), (float)(NV - 1));
      pzs = fminf(fmaxf(pzs, 0.0f), (float)(NV - 1));
      const int ix = (int)pxs;
      const int iy = (int)pys;
      const int iz = (int)pzs;

      // vol = volume[::-1] -> flip first axis; issue gather ASAP
      const int off = ((NV - 1 - ix) << 16) | (iy << 8) | iz;
      const float vox = vol[off];          // L2-resident gather (consumed 2 iters later)

      aprev = m;
      advAxis(X, m); advAxis(Y, m); advAxis(Z, m);

      // consume the 2-iterations-old load; rotate pipeline
      acc = fmaf(s0, v0, acc);
      s0 = s1; v0 = v1;
      s1 = step; v1 = vox;
    }
    // drain pipeline (order preserved -> numerically identical to serial sum)
    acc = fmaf(s0, v0, acc);
    acc = fmaf(s1, v1, acc);
  }

  const float nrm = sqrtf(sddx * sddx + sddy * sddy + sddz * sddz);
  out[i * WW + j] = acc * nrm;
}

extern "C" void kernel_launch(void* const* d_in, const int* in_sizes, int n_in,
                              void* d_out, int out_size, void* d_ws, size_t ws_size,
                              hipStream_t stream) {
  const float* vol     = (const float*)d_in[0];
  const float* spacing = (const float*)d_in[1];
  const float* sdr     = (const float*)d_in[2];
  const float* theta   = (const float*)d_in[3];
  const float* phi     = (const float*)d_in[4];
  const float* gamma   = (const float*)d_in[5];
  const float* bx      = (const float*)d_in[6];
  const float* by      = (const float*)d_in[7];
  const float* bz      = (const float*)d_in[8];
  float* out = (float*)d_out;

  // Warm L2 (64MB volume fits in 192MB L2) via global_prefetch_b8.
  const int nvol = in_sizes[0];
  const int pthreads = (nvol + 31) / 32;
  prefetch_l2<<<(pthreads + 255) / 256, 256, 0, stream>>>(vol, nvol);

  dim3 grid(WW / 8, HH / 8);   // 25 x 25 tiles of 8x8 pixels
  drr_siddon<<<grid, 64, 0, stream>>>(vol, spacing, sdr, theta, phi, gamma,
                                      bx, by, bz, out);
}